// QuantumDQN_22849226015055
// MI455X (gfx1250) — compile-verified
//
#include <hip/hip_runtime.h>
#include <math.h>

typedef float v2f __attribute__((ext_vector_type(2)));
typedef float v8f __attribute__((ext_vector_type(8)));

#define PI_F 3.14159265358979f

// CNOT as index involution: if control bit set, flip target bit.
// wire w <-> bit (3-w) of the 4-bit state index (wire 0 = MSB).
__device__ __forceinline__ int cnot_perm(int i, int c, int t) {
    if ((i >> (3 - c)) & 1) i ^= (1 << (3 - t));
    return i;
}

// Build per-layer matrices W_l = P_cnot * (tensor_prod RZ(pz_w)*RY(py_w)),
// with rows pre-permuted by tau so the WMMA D (C-layout) directly equals the
// next layer's B-operand layout.  mats layout: [layer][3][16][16] floats,
// mat 0 = Re(W), mat 1 = -Im(W), mat 2 = Im(W).
__global__ void qdqn_setup(const float* __restrict__ qp, float* __restrict__ mats) {
    int tid = threadIdx.x;       // 256 threads: one per (m,k)
    int m = tid >> 4;            // D-position row (v + 8h)
    int k = tid & 15;            // input state column
    // tau(m): state row that D position m must hold (B-layout row 4q+2h+r)
    int v = m & 7, hh = m >> 3;
    int i_out = 4 * (v >> 1) + 2 * hh + (v & 1);
    // sigma: CNOT ring [(2,3),(1,2),(0,1),(3,0)]; psi_out[i] = psi_mid[sigma(i)]
    int i_pre = i_out;
    i_pre = cnot_perm(i_pre, 3, 0);
    i_pre = cnot_perm(i_pre, 0, 1);
    i_pre = cnot_perm(i_pre, 1, 2);
    i_pre = cnot_perm(i_pre, 2, 3);

    for (int l = 0; l < 4; ++l) {
        const float* p = qp + 8 * l;
        float ar = 1.0f, ai = 0.0f;   // complex product accumulator
        #pragma unroll
        for (int w = 0; w < 4; ++w) {
            float sy, cy, sz, cz;
            __sincosf(0.5f * p[w],     &sy, &cy);   // RY(py)
            __sincosf(0.5f * p[4 + w], &sz, &cz);   // RZ(pz)
            int ib = (i_pre >> (3 - w)) & 1;
            int jb = (k     >> (3 - w)) & 1;
            // A = RZ*RY: [[cy e^{-iz/2}, -sy e^{-iz/2}],[sy e^{+iz/2}, cy e^{+iz/2}]]
            float mag = (ib == jb) ? cy : ((ib == 1) ? sy : -sy);
            float er = cz;
            float ei = (ib == 0) ? -sz : sz;
            float xr = mag * er, xi = mag * ei;
            float nr = ar * xr - ai * xi;
            float ni = ar * xi + ai * xr;
            ar = nr; ai = ni;
        }
        float* base = mats + l * 768;
        base[tid]       = ar;     // Wr
        base[256 + tid] = -ai;    // -Wi
        base[512 + tid] = ai;     // Wi
    }
}

// In-lane RX: partner row is idx^msk within this lane's 8 register slots.
#define RX_INLANE(ci, si, msk)                                                  \
    do {                                                                        \
        float nr_[8], ni_[8];                                                   \
        _Pragma("unroll") for (int i2 = 0; i2 < 8; ++i2) {                      \
            int p_ = i2 ^ (msk);                                                \
            nr_[i2] = (ci) * re[i2] + (si) * im[p_];                            \
            ni_[i2] = (ci) * im[i2] - (si) * re[p_];                            \
        }                                                                       \
        _Pragma("unroll") for (int i2 = 0; i2 < 8; ++i2) {                      \
            re[i2] = nr_[i2]; im[i2] = ni_[i2];                                 \
        }                                                                       \
    } while (0)

// Cross-half RX (wire 2): partner row lives in the other half-wave.
#define RX_CROSS(ci, si)                                                        \
    do {                                                                        \
        float pr_[8], pi_[8];                                                   \
        _Pragma("unroll") for (int i2 = 0; i2 < 8; ++i2) {                      \
            pr_[i2] = __shfl_xor(re[i2], 16, 32);                               \
            pi_[i2] = __shfl_xor(im[i2], 16, 32);                               \
        }                                                                       \
        _Pragma("unroll") for (int i2 = 0; i2 < 8; ++i2) {                      \
            float nr_ = (ci) * re[i2] + (si) * pi_[i2];                         \
            float ni_ = (ci) * im[i2] - (si) * pr_[i2];                         \
            re[i2] = nr_; im[i2] = ni_;                                         \
        }                                                                       \
    } while (0)

__global__ void __launch_bounds__(256) qdqn_main(
    const float* __restrict__ x, const float* __restrict__ w_emb,
    const float* __restrict__ mats, const float* __restrict__ Wlin,
    const float* __restrict__ bias, float* __restrict__ out, int B)
{
    const int lane = threadIdx.x & 31;
    const int wave = blockIdx.x * (blockDim.x >> 5) + (threadIdx.x >> 5);
    const int bcol = lane & 15;           // batch column within tile / A-row M
    const int h    = lane >> 4;           // half-wave
    const int batch = wave * 16 + bcol;
    if (wave * 16 >= B) return;           // wave-uniform exit keeps EXEC full
    const int bld = (batch < B) ? batch : (B - 1);

    // xe = tanh(w_emb * x) * pi ; precompute cos/sin of theta/2 per wire
    float4 xv = *(const float4*)(x + (size_t)bld * 4);
    float c[4], s[4];
    {
        float we[4]  = {w_emb[0], w_emb[1], w_emb[2], w_emb[3]};
        float xin[4] = {xv.x, xv.y, xv.z, xv.w};
        #pragma unroll
        for (int w = 0; w < 4; ++w) {
            float theta = tanhf(we[w] * xin[w]) * PI_F;
            __sincosf(0.5f * theta, &s[w], &c[w]);
        }
    }

    // State in B-operand layout: slot idx, half h holds state row
    // j = 4*(idx>>1) + 2*h + (idx&1).  |0000> -> row 0 -> idx 0, h==0.
    float re[8], im[8];
    #pragma unroll
    for (int i = 0; i < 8; ++i) { re[i] = 0.0f; im[i] = 0.0f; }
    re[0] = (h == 0) ? 1.0f : 0.0f;

    const int fo = bcol * 16 + (h << 1);  // A-frag offset: M*16 + 2h

    #pragma unroll
    for (int l = 0; l < 4; ++l) {
        // ---- batch-dependent RX gates ----
        RX_INLANE(c[0], s[0], 4);   // wire 0: row mask 8 -> idx^4
        RX_INLANE(c[1], s[1], 2);   // wire 1: row mask 4 -> idx^2
        RX_CROSS (c[2], s[2]);      // wire 2: row mask 2 -> other half-wave
        RX_INLANE(c[3], s[3], 1);   // wire 3: row mask 1 -> idx^1

        // ---- shared 16x16 complex unitary via FP32 WMMA ----
        const float* mb = mats + l * 768;
        v2f fwr[4], fwni[4], fwi[4];
        #pragma unroll
        for (int q = 0; q < 4; ++q) {
            fwr [q] = *(const v2f*)(mb +       fo + 4 * q);
            fwni[q] = *(const v2f*)(mb + 256 + fo + 4 * q);
            fwi [q] = *(const v2f*)(mb + 512 + fo + 4 * q);
        }
        v8f accr = {0.f,0.f,0.f,0.f,0.f,0.f,0.f,0.f};
        v8f acci = {0.f,0.f,0.f,0.f,0.f,0.f,0.f,0.f};
        #pragma unroll
        for (int q = 0; q < 4; ++q) {
            v2f bre; bre.x = re[2*q]; bre.y = re[2*q + 1];
            v2f bim; bim.x = im[2*q]; bim.y = im[2*q + 1];
            // Re' = Wr*Re + (-Wi)*Im ; Im' = Wi*Re + Wr*Im
            accr = __builtin_amdgcn_wmma_f32_16x16x4_f32(false, fwr[q],  false, bre, (short)0, accr, false, false);
            accr = __builtin_amdgcn_wmma_f32_16x16x4_f32(false, fwni[q], false, bim, (short)0, accr, false, false);
            acci = __builtin_amdgcn_wmma_f32_16x16x4_f32(false, fwi[q],  false, bre, (short)0, acci, false, false);
            acci = __builtin_amdgcn_wmma_f32_16x16x4_f32(false, fwr[q],  false, bim, (short)0, acci, false, false);
        }
        // Row permutation tau baked into mats => D regs ARE next-layer B regs.
        #pragma unroll
        for (int i2 = 0; i2 < 8; ++i2) { re[i2] = accr[i2]; im[i2] = acci[i2]; }
    }

    // ---- <Z_w> expectation values ----
    float ev[4];
    #pragma unroll
    for (int w = 0; w < 4; ++w) {
        float partial = 0.0f;
        #pragma unroll
        for (int v2 = 0; v2 < 8; ++v2) {
            int j = 4 * (v2 >> 1) + 2 * h + (v2 & 1);
            float p = re[v2] * re[v2] + im[v2] * im[v2];
            partial += ((j >> (3 - w)) & 1) ? -p : p;
        }
        ev[w] = partial + __shfl_xor(partial, 16, 32);
    }

    // ---- ReLU + linear head ----
    float r0 = fmaxf(ev[0], 0.f), r1 = fmaxf(ev[1], 0.f);
    float r2 = fmaxf(ev[2], 0.f), r3 = fmaxf(ev[3], 0.f);
    if (h == 0 && batch < B) {
        float4 o;
        o.x = bias[0] + Wlin[0]  * r0 + Wlin[1]  * r1 + Wlin[2]  * r2 + Wlin[3]  * r3;
        o.y = bias[1] + Wlin[4]  * r0 + Wlin[5]  * r1 + Wlin[6]  * r2 + Wlin[7]  * r3;
        o.z = bias[2] + Wlin[8]  * r0 + Wlin[9]  * r1 + Wlin[10] * r2 + Wlin[11] * r3;
        o.w = bias[3] + Wlin[12] * r0 + Wlin[13] * r1 + Wlin[14] * r2 + Wlin[15] * r3;
        *(float4*)(out + (size_t)batch * 4) = o;
    }
}

extern "C" void kernel_launch(void* const* d_in, const int* in_sizes, int n_in,
                              void* d_out, int out_size, void* d_ws, size_t ws_size,
                              hipStream_t stream) {
    const float* x     = (const float*)d_in[0];   // (B,4)
    const float* w_emb = (const float*)d_in[1];   // (4,)
    const float* qp    = (const float*)d_in[2];   // (32,)
    const float* Wlin  = (const float*)d_in[3];   // (4,4)
    const float* bias  = (const float*)d_in[4];   // (4,)
    float* out = (float*)d_out;
    int B = in_sizes[0] / 4;

    float* mats = (float*)d_ws;                   // 4*3*256 floats = 12 KB

    qdqn_setup<<<1, 256, 0, stream>>>(qp, mats);

    int waves  = (B + 15) / 16;
    int wavesPerBlock = 8;                        // 256 threads/block = 8 wave32
    int blocks = (waves + wavesPerBlock - 1) / wavesPerBlock;
    qdqn_main<<<blocks, wavesPerBlock * 32, 0, stream>>>(
        x, w_emb, mats, Wlin, bias, out, B);
}